// GSS_63015760167640
// MI455X (gfx1250) — compile-verified
//
#include <hip/hip_runtime.h>
#include <hip/hip_bf16.h>

#define BB 8
#define KK 512
#define NN 16384
#define DD 128

typedef float v2f __attribute__((ext_vector_type(2)));
typedef float v8f __attribute__((ext_vector_type(8)));

// ---------------------------------------------------------------------------
// Kernel 1: partial[b*D+d] = max_n(tgt[b,d,n]) * w_tgt[d]
// One block per (b,d) row; 256 threads; float4 loads (64 KiB/row, coalesced).
// ---------------------------------------------------------------------------
__global__ void gss_tgtmax(const float* __restrict__ tgt,
                           const float* __restrict__ conv_w,
                           float* __restrict__ partial) {
  const int bd = blockIdx.x;            // 0 .. B*D-1
  const int d  = bd % DD;
  const float4* row = (const float4*)(tgt + (size_t)bd * NN);
  float m = -__builtin_inff();
  for (int i = threadIdx.x; i < NN / 4; i += 256) {
    float4 v = row[i];
    m = fmaxf(m, fmaxf(fmaxf(v.x, v.y), fmaxf(v.z, v.w)));
  }
  __shared__ float sm[256];
  sm[threadIdx.x] = m;
  __syncthreads();
  for (int off = 128; off > 0; off >>= 1) {
    if (threadIdx.x < off)
      sm[threadIdx.x] = fmaxf(sm[threadIdx.x], sm[threadIdx.x + off]);
    __syncthreads();
  }
  if (threadIdx.x == 0) partial[bd] = sm[0] * conv_w[DD + d];
}

// ---------------------------------------------------------------------------
// Kernel 2: tb[b] = sum_d partial[b*D+d]   (the tgt_global . w_tgt bias)
// ---------------------------------------------------------------------------
__global__ void gss_tb(const float* __restrict__ partial, float* __restrict__ tb) {
  const int b = blockIdx.x;
  __shared__ float ss[128];
  ss[threadIdx.x] = partial[b * DD + threadIdx.x];
  __syncthreads();
  for (int off = 64; off > 0; off >>= 1) {
    if (threadIdx.x < off) ss[threadIdx.x] += ss[threadIdx.x + off];
    __syncthreads();
  }
  if (threadIdx.x == 0) tb[b] = ss[0];
}

// ---------------------------------------------------------------------------
// Kernel 3: logits[b,n] = sum_d src[b,d,n]*w_src[d] + tb[b]  via WMMA f32.
// Per wave: one 16-column tile of n, K-loop over D in steps of 4 using
// V_WMMA_F32_16X16X4_F32 with chained C accumulation.
//   A (16x4): every row = w[k0..k0+3]; per ISA layout lane holds M,
//             VGPR0 = K=(0|2), VGPR1 = K=(1|3) split by half-wave.
//   B (4x16): src tile, lanes 0-15 carry N=n0..n0+15.
//   D row M=0 (VGPR0, lanes 0-15) = the 16 logits.
// ---------------------------------------------------------------------------
__global__ void gss_logits_wmma(const float* __restrict__ src,
                                const float* __restrict__ conv_w,
                                const float* __restrict__ tb,
                                float* __restrict__ logits) {
  const int wave = (blockIdx.x * blockDim.x + threadIdx.x) >> 5;
  const int lane = threadIdx.x & 31;
  const int tile = wave;                      // 0 .. B*(N/16)-1
  const int b    = tile / (NN / 16);
  const int n0   = (tile % (NN / 16)) * 16;
  const int half = lane >> 4;                 // 0: K=0,1  1: K=2,3
  const int col  = n0 + (lane & 15);
  const float* srcb = src + (size_t)b * DD * NN;

  v8f c = {};
  for (int k0 = 0; k0 < DD; k0 += 4) {
    const int d0 = k0 + 2 * half;
    v2f a, bm;
    a[0]  = conv_w[d0];
    a[1]  = conv_w[d0 + 1];
    bm[0] = srcb[(size_t)d0 * NN + col];
    bm[1] = srcb[(size_t)(d0 + 1) * NN + col];
    // (neg_a, A, neg_b, B, c_mod, C, reuse_a, reuse_b)
    c = __builtin_amdgcn_wmma_f32_16x16x4_f32(false, a, false, bm,
                                              (short)0, c, false, false);
  }
  if (lane < 16)
    logits[(size_t)b * NN + n0 + lane] = c[0] + tb[b];
}

// ---------------------------------------------------------------------------
// Kernel 4: per (b,k) row of length N:
//   z = (logits + gumbel(u)) / tau ; single-pass online softmax:
//   track running max m, sum s = sum exp(z - m), first-occurrence argmax.
//   y_soft@idx = 1/s_total ; scale = (1 - y) + y  (matches reference fp).
//   Outputs are one-hot gathers scaled by `scale`.
// ---------------------------------------------------------------------------
__global__ void gss_select(const float* __restrict__ gumbel,
                           const float* __restrict__ logits,
                           const float* __restrict__ temperature,
                           const float* __restrict__ points,
                           const float* __restrict__ src,
                           float* __restrict__ out_points,
                           float* __restrict__ out_emb) {
  const int bk = blockIdx.x;
  const int b  = bk / KK;
  const int k  = bk % KK;
  const int tid = threadIdx.x;

  const float4* grow = (const float4*)(gumbel + (size_t)bk * NN);
  const float4* lrow = (const float4*)(logits + (size_t)b * NN);
  const float inv_tau = 1.0f / temperature[b];
  const float eps = 1e-10f;

  float m = -__builtin_inff();
  float s = 0.0f;
  int   idx = 0;

  for (int i = 0; i < NN / 4 / 256; ++i) {
    const int q = i * 256 + tid;
    float4 u4 = grow[q];
    float4 l4 = lrow[q];
    float us[4] = {u4.x, u4.y, u4.z, u4.w};
    float ls[4] = {l4.x, l4.y, l4.z, l4.w};
#pragma unroll
    for (int j = 0; j < 4; ++j) {
      float u = fminf(fmaxf(us[j], eps), 1.0f - eps);
      float g = -__logf(-__logf(u));
      float z = (ls[j] + g) * inv_tau;
      if (z > m) {                 // strict > keeps first occurrence
        s = s * __expf(m - z) + 1.0f;
        m = z;
        idx = q * 4 + j;
      } else {
        s += __expf(z - m);
      }
    }
  }

  __shared__ float ms[256];
  __shared__ float ssh[256];
  __shared__ int   ish[256];
  ms[tid] = m; ssh[tid] = s; ish[tid] = idx;
  __syncthreads();
  for (int off = 128; off > 0; off >>= 1) {
    if (tid < off) {
      float m1 = ms[tid],       s1 = ssh[tid];       int i1 = ish[tid];
      float m2 = ms[tid + off], s2 = ssh[tid + off]; int i2 = ish[tid + off];
      float mo = fmaxf(m1, m2);
      ssh[tid] = s1 * __expf(m1 - mo) + s2 * __expf(m2 - mo);
      ms[tid]  = mo;
      ish[tid] = (m1 > m2) ? i1 : ((m2 > m1) ? i2 : min(i1, i2));
    }
    __syncthreads();
  }

  __shared__ float scale_sh;
  __shared__ int   idx_sh;
  if (tid == 0) {
    float y  = 1.0f / ssh[0];          // y_soft at the argmax
    scale_sh = (1.0f - y) + y;         // scores value at argmax (reference fp)
    idx_sh   = ish[0];
  }
  __syncthreads();
  const float sc = scale_sh;
  const int   id = idx_sh;

  if (tid < DD) {
    out_emb[((size_t)b * DD + tid) * KK + k] =
        src[((size_t)b * DD + tid) * NN + id] * sc;
  } else if (tid < DD + 3) {
    const int c = tid - DD;
    out_points[((size_t)b * 3 + c) * KK + k] =
        points[((size_t)b * 3 + c) * NN + id] * sc;
  }
}

// ---------------------------------------------------------------------------
extern "C" void kernel_launch(void* const* d_in, const int* in_sizes, int n_in,
                              void* d_out, int out_size, void* d_ws, size_t ws_size,
                              hipStream_t stream) {
  const float* points = (const float*)d_in[0];   // (B,3,N)
  const float* src    = (const float*)d_in[1];   // (B,D,N)
  const float* tgt    = (const float*)d_in[2];   // (B,D,N)
  const float* temp   = (const float*)d_in[3];   // (B,)
  const float* conv_w = (const float*)d_in[4];   // (2D,)
  const float* gumbel = (const float*)d_in[5];   // (B,K,N)

  float* ws      = (float*)d_ws;
  float* logits  = ws;                                  // B*N floats
  float* partial = ws + (size_t)BB * NN;                // B*D floats
  float* tb      = partial + BB * DD;                   // B floats

  float* outp = (float*)d_out;                          // (B,3,K)
  float* oute = outp + (size_t)BB * 3 * KK;             // (B,D,K)

  gss_tgtmax<<<BB * DD, 256, 0, stream>>>(tgt, conv_w, partial);
  gss_tb<<<BB, 128, 0, stream>>>(partial, tb);
  gss_logits_wmma<<<(BB * NN / 16) / 8, 256, 0, stream>>>(src, conv_w, tb, logits);
  gss_select<<<BB * KK, 256, 0, stream>>>(gumbel, logits, temp, points, src, outp, oute);
}